// MultiScaleGraphNetwork_46420006535627
// MI455X (gfx1250) — compile-verified
//
#include <hip/hip_runtime.h>
#include <math.h>

#define BB   1024
#define NN   256
#define FF   64
#define FSS  128
#define OUTD 128
#define SSEG 4
#define HH   4
#define HD   16
#define KNbr 4
#define BN   (BB*NN)   // 262144
#define BS   (BB*SSEG) // 4096
#define EPSF 1e-5f
#define BIGF 1e9f

typedef float v2f __attribute__((ext_vector_type(2)));
typedef float v8f __attribute__((ext_vector_type(8)));

// ---------------------------------------------------------------------------
// Kernel 0: fold q = sq@wq+bq into score coefficients
//   A[hs][f] = (1/4) * sum_d wk[f, h*16+d] * q[s, h*16+d],  c[hs] = (1/4)*q.bk
// so scores[b,h,s,n] = x[b,n,:]·A[hs,:] + c[hs]
// ---------------------------------------------------------------------------
__global__ void k_prep(const float* __restrict__ sq, const float* __restrict__ wq,
                       const float* __restrict__ bq, const float* __restrict__ wk,
                       const float* __restrict__ bk, float* __restrict__ Acoef,
                       float* __restrict__ cvec) {
  __shared__ float qs[SSEG][FF];
  int tid = threadIdx.x;
  int s = tid >> 6, d = tid & 63;
  float acc = bq[d];
  for (int f = 0; f < FF; ++f) acc += sq[s*FF + f] * wq[f*FF + d];
  qs[s][d] = acc;
  __syncthreads();
  for (int it = 0; it < 4; ++it) {
    int idx = tid + it*256;          // 16*64 entries
    int hs = idx >> 6, f = idx & 63;
    int h = hs >> 2, s2 = hs & 3;
    float a = 0.f;
    for (int dd = 0; dd < HD; ++dd) a += wk[f*FF + h*HD + dd] * qs[s2][h*HD + dd];
    Acoef[idx] = 0.25f * a;
  }
  if (tid < 16) {
    int h = tid >> 2, s2 = tid & 3;
    float cc = 0.f;
    for (int dd = 0; dd < HD; ++dd) cc += qs[s2][h*HD + dd] * bk[h*HD + dd];
    cvec[tid] = 0.25f * cc;
  }
}

// ---------------------------------------------------------------------------
// Kernel 1: per-batch attention -> assign (B,N,S) and sf (B*S, FS)
// one 256-thread block per batch b; thread n owns node n.
// ---------------------------------------------------------------------------
__global__ void k_assign_sf(const float* __restrict__ x, const float* __restrict__ Acoef,
                            const float* __restrict__ cvec, const float* __restrict__ w_a2s,
                            const float* __restrict__ b_a2s, float* __restrict__ assign_out,
                            float* __restrict__ sf_out) {
  __shared__ float As[16][FF];
  __shared__ float cs[16];
  __shared__ float scr[16][NN + 1];
  __shared__ float mxs[16], sums[16];
  __shared__ float asg[NN][SSEG];
  __shared__ float ts[SSEG][FF];
  __shared__ float msum[SSEG];

  const int b = blockIdx.x, tid = threadIdx.x;
  for (int idx = tid; idx < 16*FF; idx += 256) As[idx >> 6][idx & 63] = Acoef[idx];
  if (tid < 16) cs[tid] = cvec[tid];
  __syncthreads();

  // scores for this node, all 16 (h,s) combos
  const float* xr = x + ((size_t)b*NN + tid) * FF;
  float sc[16];
  #pragma unroll
  for (int hs = 0; hs < 16; ++hs) sc[hs] = cs[hs];
  for (int f = 0; f < FF; ++f) {
    float xv = xr[f];
    #pragma unroll
    for (int hs = 0; hs < 16; ++hs) sc[hs] += xv * As[hs][f];
  }
  #pragma unroll
  for (int hs = 0; hs < 16; ++hs) scr[hs][tid] = sc[hs];
  __syncthreads();

  if (tid < 16) {                       // softmax denominators over n
    float mx = -3e38f;
    for (int n = 0; n < NN; ++n) mx = fmaxf(mx, scr[tid][n]);
    float sm = 0.f;
    for (int n = 0; n < NN; ++n) sm += expf(scr[tid][n] - mx);
    mxs[tid] = mx; sums[tid] = sm;
  }
  __syncthreads();

  // attn (mean over h) then assign = softmax over s
  float at[SSEG];
  #pragma unroll
  for (int s = 0; s < SSEG; ++s) {
    float a = 0.f;
    #pragma unroll
    for (int h = 0; h < HH; ++h) {
      int hs = h*SSEG + s;
      a += expf(sc[hs] - mxs[hs]) / sums[hs];
    }
    at[s] = 0.25f * a;
  }
  float am = fmaxf(fmaxf(at[0], at[1]), fmaxf(at[2], at[3]));
  float e[SSEG], es = 0.f;
  #pragma unroll
  for (int s = 0; s < SSEG; ++s) { e[s] = expf(at[s] - am); es += e[s]; }
  #pragma unroll
  for (int s = 0; s < SSEG; ++s) {
    float a = e[s] / es;
    asg[tid][s] = a;
    assign_out[((size_t)b*NN + tid)*SSEG + s] = a;
  }
  __syncthreads();

  // t[s][g] = sum_n assign[n,s]*x[n,g]  (thread = (s,g))
  {
    int s = tid >> 6, g = tid & 63;
    float acc = 0.f;
    for (int n = 0; n < NN; ++n)
      acc += asg[n][s] * x[((size_t)b*NN + n)*FF + g];
    ts[s][g] = acc;
  }
  if (tid < SSEG) {
    float ms = 0.f;
    for (int n = 0; n < NN; ++n) ms += asg[n][tid];
    msum[tid] = ms;
  }
  __syncthreads();

  // sf[b,s,:] = t[s,:] @ w_a2s + msum[s]*b_a2s  (thread = (s,j), j and j+64)
  {
    int s = tid >> 6, j = tid & 63;
    float a0 = msum[s] * b_a2s[j];
    float a1 = msum[s] * b_a2s[j + 64];
    for (int g = 0; g < FF; ++g) {
      float tv = ts[s][g];
      a0 += tv * w_a2s[g*FSS + j];
      a1 += tv * w_a2s[g*FSS + j + 64];
    }
    sf_out[((size_t)b*SSEG + s)*FSS + j]      = a0;
    sf_out[((size_t)b*SSEG + s)*FSS + j + 64] = a1;
  }
}

// ---------------------------------------------------------------------------
// WMMA fp32 GEMM: C[M x Nc] = A[M x Kd] @ B[Kd x Nc] (+ column bias)
// one wave per 16x16 tile, V_WMMA_F32_16X16X4_F32, K stepped by 4.
// Fragment layouts per CDNA5 ISA 7.12.2 (A: v0={K0|K2}, v1={K1|K3};
// B mirrored; C/D: vgpr r -> M=r (lanes 0-15) / M=r+8 (lanes 16-31)).
// ---------------------------------------------------------------------------
__global__ void wmma_gemm_f32(const float* __restrict__ A, const float* __restrict__ Bw,
                              const float* __restrict__ bias, float* __restrict__ C,
                              int M, int Kd, int Nc) {
  int wave = threadIdx.x >> 5;
  int lane = threadIdx.x & 31;
  int half = lane >> 4;
  int l16  = lane & 15;
  int m0 = (blockIdx.x*4 + wave) * 16;
  int n0 = blockIdx.y * 16;
  if (m0 >= M) return;

  v8f acc;
  float bv = bias ? bias[n0 + l16] : 0.0f;
  #pragma unroll
  for (int r = 0; r < 8; ++r) acc[r] = bv;

  const float* arow = A + (size_t)(m0 + l16) * Kd;
  for (int kk = 0; kk < Kd; kk += 4) {
    int ka = kk + 2*half;
    v2f a; a.x = arow[ka]; a.y = arow[ka + 1];
    v2f bfr;
    bfr.x = Bw[(size_t)ka * Nc + n0 + l16];
    bfr.y = Bw[(size_t)(ka + 1) * Nc + n0 + l16];
    acc = __builtin_amdgcn_wmma_f32_16x16x4_f32(false, a, false, bfr,
                                                (short)0, acc, false, false);
  }
  #pragma unroll
  for (int r = 0; r < 8; ++r) {
    int mr = m0 + r + 8*half;
    C[(size_t)mr * Nc + n0 + l16] = acc[r];
  }
}

// ---------------------------------------------------------------------------
// Coarse conv: z_i = h_i + (sum_j h_j - h_i)/3 within each 4-node group,
// accumulate BN stats (sum, sumsq per feature). block=128 threads (f), 1 group.
// ---------------------------------------------------------------------------
__global__ void k_coarse_combine(float* __restrict__ z, float* __restrict__ stats) {
  int f = threadIdx.x;
  size_t base = (size_t)blockIdx.x * SSEG * FSS;
  float h[SSEG], sum = 0.f;
  #pragma unroll
  for (int i = 0; i < SSEG; ++i) { h[i] = z[base + i*FSS + f]; sum += h[i]; }
  float s1 = 0.f, s2 = 0.f;
  #pragma unroll
  for (int i = 0; i < SSEG; ++i) {
    float zi = h[i] + (sum - h[i]) * (1.0f/3.0f);
    z[base + i*FSS + f] = zi;
    s1 += zi; s2 += zi*zi;
  }
  atomicAdd(&stats[f], s1);
  atomicAdd(&stats[FSS + f], s2);
}

// ---------------------------------------------------------------------------
// Generic batchnorm+relu normalize pass. nfMask = Nfeat-1 (power of 2).
// ---------------------------------------------------------------------------
__global__ void k_bn_norm(const float* __restrict__ z, const float* __restrict__ stats,
                          const float* __restrict__ gamma, const float* __restrict__ beta,
                          float* __restrict__ out, int nfMask, int sqOff, float invM) {
  size_t idx = (size_t)blockIdx.x * blockDim.x + threadIdx.x;
  int f = (int)(idx & (size_t)nfMask);
  float mu  = stats[f] * invM;
  float var = stats[sqOff + f] * invM - mu*mu;
  float v = (z[idx] - mu) * rsqrtf(var + EPSF) * gamma[f] + beta[f];
  out[idx] = fmaxf(v, 0.f);
}

// ---------------------------------------------------------------------------
// ctx broadcast: af[r,f] = x[r,f] + sum_s assign[r,s]*U[b*4+s,f] + b_s2a[f]
// ---------------------------------------------------------------------------
__global__ void k_ctx_add(const float* __restrict__ x, const float* __restrict__ assign,
                          const float* __restrict__ U, const float* __restrict__ b_s2a,
                          float* __restrict__ af) {
  int tid = threadIdx.x;
  size_t r = (size_t)blockIdx.x * 4 + (tid >> 6);
  int f = tid & 63;
  size_t b = r >> 8;
  float acc = x[r*FF + f] + b_s2a[f];
  const float* as = assign + r*SSEG;
  #pragma unroll
  for (int s = 0; s < SSEG; ++s) acc += as[s] * U[(b*SSEG + s)*FF + f];
  af[r*FF + f] = acc;
}

// ---------------------------------------------------------------------------
// kNN edges: per-batch block, thread i keeps sorted 5-best (incl. self at 0).
// ---------------------------------------------------------------------------
__global__ void k_knn(const float* __restrict__ pos, const float* __restrict__ pres,
                      int* __restrict__ dstIdx, float* __restrict__ em,
                      float* __restrict__ deg) {
  __shared__ float px[NN], py[NN];
  __shared__ int vld[NN];
  __shared__ int nvalid;
  int b = blockIdx.x, i = threadIdx.x;
  size_t g = (size_t)b*NN + i;
  float xv = pos[g*2], yv = pos[g*2 + 1];
  int v = pres[g] > 0.5f ? 1 : 0;
  px[i] = xv; py[i] = yv; vld[i] = v;
  if (i == 0) nvalid = 0;
  __syncthreads();
  if (v) atomicAdd(&nvalid, 1);
  __syncthreads();

  float bd[5]; int bi[5];
  #pragma unroll
  for (int m = 0; m < 5; ++m) { bd[m] = 3e38f; bi[m] = 0; }
  for (int j = 0; j < NN; ++j) {
    float dx = xv - px[j], dy = yv - py[j];
    float d2 = vld[j] ? (dx*dx + dy*dy) : BIGF;
    if (d2 < bd[4]) {                    // stable insertion (ties keep lower j first)
      int p = 4;
      while (p > 0 && d2 < bd[p-1]) { bd[p] = bd[p-1]; bi[p] = bi[p-1]; --p; }
      bd[p] = d2; bi[p] = j;
    }
  }
  bool okb = (v != 0) && (nvalid >= 2);
  #pragma unroll
  for (int k = 0; k < KNbr; ++k) {
    float nd = bd[k + 1];
    bool e = okb && (nd < 0.5f * BIGF);
    int dst = e ? (b*NN + bi[k + 1]) : 0;
    size_t eidx = g*KNbr + k;
    dstIdx[eidx] = dst;
    em[eidx] = e ? 1.0f : 0.0f;
    if (e) atomicAdd(&deg[dst], 1.0f);
  }
}

// ---------------------------------------------------------------------------
// edge scatter: agg[dst,f] += em * H[src,f]
// ---------------------------------------------------------------------------
__global__ void k_scatter(const float* __restrict__ H, const int* __restrict__ dstIdx,
                          const float* __restrict__ em, float* __restrict__ agg) {
  size_t gid = (size_t)blockIdx.x * 256 + threadIdx.x;   // BN*K*64 threads
  size_t e = gid >> 6;
  int f = (int)(gid & 63);
  float emv = em[e];
  if (emv != 0.f) {
    size_t srcN = e >> 2;                                 // edge e = node*K + k
    int dst = dstIdx[e];
    atomicAdd(&agg[(size_t)dst*FF + f], emv * H[srcN*FF + f]);
  }
}

// ---------------------------------------------------------------------------
// z = H + agg/clip(deg,1) in place, accumulate BN stats. block owns 64 rows.
// ---------------------------------------------------------------------------
__global__ void k_zstats(float* __restrict__ H, const float* __restrict__ agg,
                         const float* __restrict__ deg, float* __restrict__ stats) {
  __shared__ float ls[FF], lq[FF];
  int tid = threadIdx.x;
  int f = tid & 63, r0 = tid >> 6;
  if (tid < FF) { ls[tid] = 0.f; lq[tid] = 0.f; }
  __syncthreads();
  size_t rowbase = (size_t)blockIdx.x * 64;
  float s1 = 0.f, s2 = 0.f;
  for (int it = 0; it < 16; ++it) {
    size_t row = rowbase + r0 + it*4;
    size_t ix = row*FF + f;
    float d = deg[row]; d = d < 1.f ? 1.f : d;
    float zv = H[ix] + agg[ix] / d;
    H[ix] = zv;
    s1 += zv; s2 += zv*zv;
  }
  atomicAdd(&ls[f], s1);
  atomicAdd(&lq[f], s2);
  __syncthreads();
  if (tid < FF) {
    atomicAdd(&stats[tid], ls[tid]);
    atomicAdd(&stats[FF + tid], lq[tid]);
  }
}

// ---------------------------------------------------------------------------
// pooling + output GEMM: one block per batch.
// ---------------------------------------------------------------------------
__global__ void k_pool_out(const float* __restrict__ af, const float* __restrict__ pres,
                           const float* __restrict__ w_out, const float* __restrict__ b_out,
                           float* __restrict__ out) {
  __shared__ float part[4][FF];
  __shared__ float pp[NN];
  __shared__ float gfeat[FF];
  __shared__ float psum;
  int b = blockIdx.x, tid = threadIdx.x;
  int f = tid & 63, c = tid >> 6;
  float acc = 0.f;
  for (int n = c*64; n < c*64 + 64; ++n) {
    float p = pres[(size_t)b*NN + n];
    acc += af[((size_t)b*NN + n)*FF + f] * p;
  }
  part[c][f] = acc;
  pp[tid] = pres[(size_t)b*NN + tid];
  __syncthreads();
  if (tid == 0) {
    float s = 0.f;
    for (int n = 0; n < NN; ++n) s += pp[n];
    psum = fmaxf(s, 1e-8f);
  }
  __syncthreads();
  if (tid < FF)
    gfeat[tid] = (part[0][tid] + part[1][tid] + part[2][tid] + part[3][tid]) / psum;
  __syncthreads();
  if (tid < OUTD) {
    float o = b_out[tid];
    for (int f0 = 0; f0 < FF; ++f0) o += gfeat[f0] * w_out[f0*OUTD + tid];
    out[(size_t)b*OUTD + tid] = o;
  }
}

// ---------------------------------------------------------------------------
extern "C" void kernel_launch(void* const* d_in, const int* in_sizes, int n_in,
                              void* d_out, int out_size, void* d_ws, size_t ws_size,
                              hipStream_t stream) {
  const float* x       = (const float*)d_in[0];
  const float* pos     = (const float*)d_in[1];
  const float* pres    = (const float*)d_in[2];
  const float* sq      = (const float*)d_in[3];
  const float* wq      = (const float*)d_in[4];
  const float* bq      = (const float*)d_in[5];
  const float* wk      = (const float*)d_in[6];
  const float* bk      = (const float*)d_in[7];
  const float* w_a2s   = (const float*)d_in[8];
  const float* b_a2s   = (const float*)d_in[9];
  const float* w_s2a   = (const float*)d_in[10];
  const float* b_s2a   = (const float*)d_in[11];
  const float* coarseW = (const float*)d_in[12];
  const float* coarseB = (const float*)d_in[13];
  const float* coarseG = (const float*)d_in[14];
  const float* coarseBt= (const float*)d_in[15];
  const float* fineW   = (const float*)d_in[16];
  const float* fineB   = (const float*)d_in[17];
  const float* fineG   = (const float*)d_in[18];
  const float* fineBt  = (const float*)d_in[19];
  const float* w_out   = (const float*)d_in[20];
  const float* b_out   = (const float*)d_in[21];
  float* out = (float*)d_out;

  float* W = (float*)d_ws;
  size_t o = 0;
  float* Acoef  = W + o; o += 16*FF;          // 1024
  float* cvec   = W + o; o += 16;
  o = 1056;
  float* stats  = W + o; o += 256;
  float* assign = W + o; o += (size_t)BN*SSEG;     // 1M
  float* sfA    = W + o; o += (size_t)BS*FSS;      // 512K
  float* sfB    = W + o; o += (size_t)BS*FSS;
  float* zbuf   = W + o; o += (size_t)BS*FSS;
  float* Ubuf   = W + o; o += (size_t)BS*FF;       // 256K
  float* emb    = W + o; o += (size_t)BN*KNbr;     // 1M
  int*   dstIdx = (int*)(W + o); o += (size_t)BN*KNbr;
  float* degb   = W + o; o += (size_t)BN;          // 256K
  float* af     = W + o; o += (size_t)BN*FF;       // 16M
  float* Hb     = W + o; o += (size_t)BN*FF;
  float* aggb   = W + o; o += (size_t)BN*FF;

  // 0) score coefficients
  k_prep<<<1, 256, 0, stream>>>(sq, wq, bq, wk, bk, Acoef, cvec);

  // 1) attention -> assign + sf
  k_assign_sf<<<BB, 256, 0, stream>>>(x, Acoef, cvec, w_a2s, b_a2s, assign, sfA);

  // 2) coarse graph conv x2
  float* sin = sfA; float* sout = sfB;
  for (int l = 0; l < 2; ++l) {
    hipMemsetAsync(stats, 0, 256*sizeof(float), stream);
    wmma_gemm_f32<<<dim3(BS/64, FSS/16), 128, 0, stream>>>(
        sin, coarseW + (size_t)l*FSS*FSS, coarseB + l*FSS, zbuf, BS, FSS, FSS);
    k_coarse_combine<<<BB, FSS, 0, stream>>>(zbuf, stats);
    k_bn_norm<<<(BS*FSS)/256, 256, 0, stream>>>(
        zbuf, stats, coarseG + l*FSS, coarseBt + l*FSS, sout,
        FSS - 1, FSS, 1.0f/(float)BS);
    float* t = sin; sin = sout; sout = t;
  }
  // sin now holds final sf

  // 3) U = sf @ w_s2a  (bias applied in ctx_add)
  wmma_gemm_f32<<<dim3(BS/64, FF/16), 128, 0, stream>>>(
      sin, w_s2a, nullptr, Ubuf, BS, FSS, FF);

  // 4) af = x + assign@U + b_s2a
  k_ctx_add<<<BN/4, 256, 0, stream>>>(x, assign, Ubuf, b_s2a, af);

  // 5) kNN edges + degrees
  hipMemsetAsync(degb, 0, (size_t)BN*sizeof(float), stream);
  k_knn<<<BB, NN, 0, stream>>>(pos, pres, dstIdx, emb, degb);

  // 6) fine graph conv x3
  for (int l = 0; l < 3; ++l) {
    hipMemsetAsync(aggb, 0, (size_t)BN*FF*sizeof(float), stream);
    hipMemsetAsync(stats, 0, 256*sizeof(float), stream);
    wmma_gemm_f32<<<dim3(BN/64, FF/16), 128, 0, stream>>>(
        af, fineW + (size_t)l*FF*FF, fineB + l*FF, Hb, BN, FF, FF);
    k_scatter<<<(size_t)BN*KNbr*FF/256, 256, 0, stream>>>(Hb, dstIdx, emb, aggb);
    k_zstats<<<BN/64, 256, 0, stream>>>(Hb, aggb, degb, stats);
    k_bn_norm<<<((size_t)BN*FF)/256, 256, 0, stream>>>(
        Hb, stats, fineG + l*FF, fineBt + l*FF, af,
        FF - 1, FF, 1.0f/(float)BN);
  }

  // 7) pooling + output projection
  k_pool_out<<<BB, 256, 0, stream>>>(af, pres, w_out, b_out, out);

  (void)in_sizes; (void)n_in; (void)out_size; (void)ws_size;
}